// DynamicFilterLayerOneChannel_58780922413139
// MI455X (gfx1250) — compile-verified
//
#include <hip/hip_runtime.h>

// Dynamic per-pixel filtering (N=4, C=1, H=544, W=960, 9x9 window, same pad).
// Bandwidth-bound on `filters` (~677 MB streamed once). Strategy:
//  - filters: 128-bit coalesced non-temporal loads (no reuse -> TH=NT)
//  - x_in: staged into LDS (with halo) via CDNA5 async global->LDS ops
//  - 9x9 fully-unrolled FMA with LDS b128 sliding-window reads

constexpr int DFL_K   = 9;
constexpr int DFL_PAD = 4;
constexpr int DFL_N   = 4;
constexpr int DFL_H   = 544;
constexpr int DFL_W   = 960;
constexpr long long DFL_HW = (long long)DFL_H * DFL_W;

constexpr int TILE_W  = 64;   // 16 threads x 4 pixels each
constexpr int TILE_H  = 16;
constexpr int SM_COLS = TILE_W + 2 * DFL_PAD;  // 72 floats (288 B, 16B aligned rows)
constexpr int SM_ROWS = TILE_H + 2 * DFL_PAD;  // 24
constexpr int NTHREADS = 256;                  // 8 wave32s

typedef float v4f __attribute__((ext_vector_type(4)));

#define AS_GLOBAL __attribute__((address_space(1)))
#define AS_LOCAL  __attribute__((address_space(3)))

// --- CDNA5 async global->LDS staging path selection --------------------------
#if __has_builtin(__builtin_amdgcn_global_load_async_to_lds_b32)
#define DFL_ASYNC_MODE 1
#elif __has_builtin(__builtin_amdgcn_global_load_lds)
#define DFL_ASYNC_MODE 2
#else
#define DFL_ASYNC_MODE 0
#endif

__device__ __forceinline__ void dfl_async_copy_b32(const float* gsrc, float* ldst) {
#if DFL_ASYNC_MODE == 1
  // Prototype (from hipcc diagnostic): (global int*, local int*, imm offset, imm cpol)
  __builtin_amdgcn_global_load_async_to_lds_b32(
      (AS_GLOBAL int*)gsrc, (AS_LOCAL int*)ldst, 0, 0);
#elif DFL_ASYNC_MODE == 2
  __builtin_amdgcn_global_load_lds(
      (AS_GLOBAL const unsigned int*)gsrc, (AS_LOCAL unsigned int*)ldst, 4, 0, 0);
#else
  *ldst = *gsrc;
#endif
}

__device__ __forceinline__ void dfl_async_wait() {
#if DFL_ASYNC_MODE != 0
#if __has_builtin(__builtin_amdgcn_s_wait_asynccnt)
  __builtin_amdgcn_s_wait_asynccnt(0);
#else
  asm volatile("s_wait_asynccnt 0" ::: "memory");
#endif
#endif
}
// -----------------------------------------------------------------------------

__global__ __launch_bounds__(NTHREADS) void
DynamicFilterLayerOneChannel_58780922413139_kernel(const float* __restrict__ x,
                                                   const float* __restrict__ filt,
                                                   const float* __restrict__ bias,
                                                   float* __restrict__ out) {
  __shared__ __attribute__((aligned(16))) float smem[SM_ROWS * SM_COLS];

  const int tid = threadIdx.x;
  const int n   = blockIdx.z;
  const int h0  = blockIdx.y * TILE_H;
  const int w0t = blockIdx.x * TILE_W;

  // ---- Stage x_in tile (+4-pixel halo, zero padded) into LDS ----
  const float* xn = x + (long long)n * DFL_HW;
#pragma unroll
  for (int idx = tid; idx < SM_ROWS * SM_COLS; idx += NTHREADS) {
    const int r  = idx / SM_COLS;
    const int c  = idx - r * SM_COLS;
    const int gh = h0 + r - DFL_PAD;
    const int gw = w0t + c - DFL_PAD;
    if ((unsigned)gh < (unsigned)DFL_H && (unsigned)gw < (unsigned)DFL_W) {
      dfl_async_copy_b32(xn + (long long)gh * DFL_W + gw, &smem[idx]);
    } else {
      smem[idx] = 0.0f;  // zero "same" padding
    }
  }
  dfl_async_wait();   // own async writes to LDS complete
  __syncthreads();    // all waves' writes visible

  // ---- Compute: each thread owns 4 consecutive w pixels of one row ----
  const int tx = tid & 15;        // 0..15 -> column group
  const int ty = tid >> 4;        // 0..15 -> row in tile
  const int h  = h0 + ty;
  const int w  = w0t + (tx << 2);

  const long long pix   = (long long)n * DFL_HW + (long long)h * DFL_W + w;
  const float*    fbase = filt + (long long)n * (DFL_K * DFL_K) * DFL_HW
                               + (long long)h * DFL_W + w;

  __builtin_prefetch(fbase, 0, 0);  // gfx1250 global_prefetch_b8, warm the stream

  v4f acc = __builtin_nontemporal_load((const v4f*)(bias + pix));

#pragma unroll
  for (int ky = 0; ky < DFL_K; ++ky) {
    // 12-float sliding window for this row: covers kx 0..8 for 4 pixels
    const float* sr = &smem[(ty + ky) * SM_COLS + (tx << 2)];
    const v4f a0 = *(const v4f*)(sr);
    const v4f a1 = *(const v4f*)(sr + 4);
    const v4f a2 = *(const v4f*)(sr + 8);
    const float xv[12] = {a0.x, a0.y, a0.z, a0.w,
                          a1.x, a1.y, a1.z, a1.w,
                          a2.x, a2.y, a2.z, a2.w};

    const float* frow = fbase + (long long)(ky * DFL_K) * DFL_HW;
#pragma unroll
    for (int kx = 0; kx < DFL_K; ++kx) {
      // one coalesced, non-temporal 128-bit filter load per tap per lane
      const v4f f = __builtin_nontemporal_load(
          (const v4f*)(frow + (long long)kx * DFL_HW));
      acc.x = fmaf(xv[kx + 0], f.x, acc.x);
      acc.y = fmaf(xv[kx + 1], f.y, acc.y);
      acc.z = fmaf(xv[kx + 2], f.z, acc.z);
      acc.w = fmaf(xv[kx + 3], f.w, acc.w);
    }
  }

  __builtin_nontemporal_store(acc, (v4f*)(out + pix));
}

extern "C" void kernel_launch(void* const* d_in, const int* in_sizes, int n_in,
                              void* d_out, int out_size, void* d_ws, size_t ws_size,
                              hipStream_t stream) {
  const float* x    = (const float*)d_in[0];  // [N,1,H,W]
  const float* filt = (const float*)d_in[1];  // [N,81,H,W]
  const float* bias = (const float*)d_in[2];  // [N,1,H,W]
  float*       out  = (float*)d_out;          // [N,1,H,W]

  dim3 grid(DFL_W / TILE_W,   // 15
            DFL_H / TILE_H,   // 34
            DFL_N);           // 4
  DynamicFilterLayerOneChannel_58780922413139_kernel<<<grid, NTHREADS, 0, stream>>>(
      x, filt, bias, out);
}